// YOLOlossVectorized_61804579389966
// MI455X (gfx1250) — compile-verified
//
#include <hip/hip_runtime.h>

// ---------------------------------------------------------------------------
// YOLO loss on MI455X (gfx1250).
// HBM-bound streaming reduction: 176 MB in, 4 bytes out (~7.6 us floor @ 23.3 TB/s).
// CDNA5 specifics used:
//   * wave32 reduction via V_WMMA_F32_16X16X4_F32 (B = ones => row-sum engine)
//   * float2 (b64) vector loads on the 8B-aligned prediction stream
//   * deterministic two-pass reduction (no float atomics)
// ---------------------------------------------------------------------------

typedef float v2f __attribute__((ext_vector_type(2)));
typedef float v8f __attribute__((ext_vector_type(8)));

#define YS 7
#define YC 20
#define PRED_STRIDE 30   // C + B*5
#define TGT_STRIDE  25   // C + 5
#define LAMBDA_COORD 5.0f
#define LAMBDA_NOOBJ 0.5f
#define YEPS 1e-6f

// Exact f32 sum of all 32 lanes of the wave using one WMMA.
// A (16x4 f32) holds each lane's value once (other A slots = 0), B = ones.
// D[m][n] = sum_k A[m][k]; summing the 8 C/D VGPRs on lane n gives rows 0..7
// (lanes 0-15) or rows 8..15 (lanes 16-31) at column n; xor-16 pairs them so
// every lane ends with sum_m sum_k A[m][k] = sum of all lane inputs.
// Products are v*1.0 (exact), accumulation in f32 -> full precision.
// EXEC must be all 1s at the call site (callers guarantee convergence).
__device__ __forceinline__ float wave_sum32(float v) {
    v2f a; a[0] = v;    a[1] = 0.0f;
    v2f b; b[0] = 1.0f; b[1] = 1.0f;
    v8f c = {};
    v8f d = __builtin_amdgcn_wmma_f32_16x16x4_f32(
        /*neg_a=*/false, a, /*neg_b=*/false, b,
        /*c_mod=*/(short)0, c, /*reuse_a=*/false, /*reuse_b=*/false);
    float s = d[0] + d[1] + d[2] + d[3] + d[4] + d[5] + d[6] + d[7];
    s += __shfl_xor(s, 16, 32);
    return s;
}

__device__ __forceinline__ float iou_fn(float ax, float ay, float aw, float ah,
                                        float bx, float by, float bw, float bh) {
    float ax1 = ax - aw * 0.5f, ax2 = ax + aw * 0.5f;
    float ay1 = ay - ah * 0.5f, ay2 = ay + ah * 0.5f;
    float bx1 = bx - bw * 0.5f, bx2 = bx + bw * 0.5f;
    float by1 = by - bh * 0.5f, by2 = by + bh * 0.5f;
    float iw = fmaxf(fminf(ax2, bx2) - fmaxf(ax1, bx1), 0.0f);
    float ih = fmaxf(fminf(ay2, by2) - fmaxf(ay1, by1), 0.0f);
    float inter = iw * ih;
    float uni = (ax2 - ax1) * (ay2 - ay1) + (bx2 - bx1) * (by2 - by1) - inter;
    return inter / (uni + YEPS);
}

__global__ void __launch_bounds__(256)
yolo_cell_kernel(const float* __restrict__ pred,
                 const float* __restrict__ tgt,
                 float* __restrict__ block_partials,
                 int ncells) {
    const int cell = blockIdx.x * 256 + threadIdx.x;
    float loss = 0.0f;

    if (cell < ncells) {
        const float* p = pred + (size_t)cell * PRED_STRIDE;  // 120B stride: 8B aligned
        const float* t = tgt  + (size_t)cell * TGT_STRIDE;   // 100B stride: 4B aligned
        const v2f*   p2 = (const v2f*)p;                     // global_load_b64 path

        // Class SSE: preds[0..19] vs targets[0..19], accumulated streaming.
        float cls = 0.0f;
#pragma unroll
        for (int i = 0; i < 10; ++i) {
            v2f pc = p2[i];
            float d0 = pc[0] - t[2 * i];
            float d1 = pc[1] - t[2 * i + 1];
            cls = fmaf(d0, d0, cls);
            cls = fmaf(d1, d1, cls);
        }

        // box1 = p[20..24], box2 = p[25..29]  (p2[10..14])
        float bx[10];
#pragma unroll
        for (int i = 0; i < 5; ++i) {
            v2f v = p2[10 + i];
            bx[2 * i]     = v[0];
            bx[2 * i + 1] = v[1];
        }

        // tbox = targets[20..24]; NOTE: tbox[0] is the object flag (reference quirk).
        float tb0 = t[YC + 0], tb1 = t[YC + 1], tb2 = t[YC + 2],
              tb3 = t[YC + 3], tb4 = t[YC + 4];

        float iou1 = iou_fn(bx[0], bx[1], bx[2], bx[3], tb0, tb1, tb2, tb3);
        float iou2 = iou_fn(bx[5], bx[6], bx[7], bx[8], tb0, tb1, tb2, tb3);
        bool sel = iou1 > iou2;

        float r0  = sel ? bx[0] : bx[5];
        float r1  = sel ? bx[1] : bx[6];
        float r2  = sel ? bx[2] : bx[7];
        float r3  = sel ? bx[3] : bx[8];
        float r4  = sel ? bx[4] : bx[9];
        float nr4 = sel ? bx[9] : bx[4];

        float d0 = r0 - tb0;
        float d1 = r1 - tb1;
        float sw = sqrtf(fmaxf(r2, YEPS)) - sqrtf(fmaxf(tb2, YEPS));
        float sh = sqrtf(fmaxf(r3, YEPS)) - sqrtf(fmaxf(tb3, YEPS));
        float coord = LAMBDA_COORD * (d0 * d0 + d1 * d1 + sw * sw + sh * sh);

        float dc = r4 - tb4;
        float obj_loss   = coord + dc * dc + cls + LAMBDA_NOOBJ * nr4 * nr4;
        float noobj_loss = LAMBDA_NOOBJ * (bx[4] * bx[4] + bx[9] * bx[9]);

        loss = (tb0 == 1.0f) ? obj_loss : noobj_loss;
    }

    // Converged here: EXEC all-1s for the WMMA reduction.
    float ws = wave_sum32(loss);

    __shared__ float wsum[8];
    if ((threadIdx.x & 31) == 0) wsum[threadIdx.x >> 5] = ws;
    __syncthreads();
    if (threadIdx.x == 0) {
        float bs = 0.0f;
#pragma unroll
        for (int i = 0; i < 8; ++i) bs += wsum[i];
        block_partials[blockIdx.x] = bs;
    }
}

__global__ void __launch_bounds__(256)
yolo_final_reduce(const float* __restrict__ partials, int n,
                  float* __restrict__ out, float inv_batch) {
    float s = 0.0f;
    for (int i = threadIdx.x; i < n; i += 256) s += partials[i];

    float ws = wave_sum32(s);

    __shared__ float wsum[8];
    if ((threadIdx.x & 31) == 0) wsum[threadIdx.x >> 5] = ws;
    __syncthreads();
    if (threadIdx.x == 0) {
        float bs = 0.0f;
#pragma unroll
        for (int i = 0; i < 8; ++i) bs += wsum[i];
        out[0] = bs * inv_batch;
    }
}

extern "C" void kernel_launch(void* const* d_in, const int* in_sizes, int n_in,
                              void* d_out, int out_size, void* d_ws, size_t ws_size,
                              hipStream_t stream) {
    const float* pred = (const float*)d_in[0];
    const float* tgt  = (const float*)d_in[1];

    const int batch  = in_sizes[0] / (YS * YS * PRED_STRIDE);  // 16384
    const int ncells = batch * YS * YS;                        // 802816
    const int blocks = (ncells + 255) / 256;                   // 3136

    float* partials = (float*)d_ws;  // blocks * 4 bytes of scratch

    yolo_cell_kernel<<<blocks, 256, 0, stream>>>(pred, tgt, partials, ncells);
    yolo_final_reduce<<<1, 256, 0, stream>>>(partials, blocks, (float*)d_out,
                                             1.0f / (float)batch);
}